// Aggregator_14508399526387
// MI455X (gfx1250) — compile-verified
//
#include <hip/hip_runtime.h>

// GraphSAGE aggregator, fused: mean-over-neighbors + dual GEMM + bias + ReLU.
// Bandwidth-bound (~1.17 GB traffic @ 23.3 TB/s -> ~50us floor); GEMM done with
// full-precision V_WMMA_F32_16X16X4_F32 so math matches the f32 reference.

typedef __attribute__((ext_vector_type(2))) float v2f;
typedef __attribute__((ext_vector_type(8))) float v8f;

#define MROWS 16    // rows of (B*H) per workgroup
#define FDIM  128   // feature dim (K)
#define NNEI  25    // neighbors
#define DHALF 128   // output columns per half
#define DOUT  256   // total output columns
#define LDA   132   // LDS row pitch (pad: 132 mod 64 = 4 -> 16 rows hit 16 banks)

__global__ __launch_bounds__(256) void sage_wmma_kernel(
    const float* __restrict__ x_self,   // (81920, 128)
    const float* __restrict__ x_neigh,  // (81920, 25, 128)
    const float* __restrict__ w_neigh,  // (128, 128)
    const float* __restrict__ w_self,   // (128, 128)
    const float* __restrict__ bias,     // (256,)
    float* __restrict__ out)            // (81920, 256)
{
    __shared__ float As[MROWS][LDA];   // x_self tile
    __shared__ float An[MROWS][LDA];   // mean(x_neigh) tile

    const int tid = threadIdx.x;
    const size_t m0 = (size_t)blockIdx.x * MROWS;

    // ---------------- Phase 1: cooperative stream + mean ----------------
    // 16 rows x 32 float4 groups = 512 slots; 256 threads -> 2 slots each.
    {
        const float4* xs = reinterpret_cast<const float4*>(x_self);
        const float4* xn = reinterpret_cast<const float4*>(x_neigh);
        const float inv_n = 1.0f / (float)NNEI;
        #pragma unroll
        for (int it = 0; it < 2; ++it) {
            const int s   = tid + it * 256;
            const int row = s >> 5;
            const int grp = s & 31;

            // x_self tile: one float4 per slot
            float4 sv = xs[(m0 + row) * (FDIM / 4) + grp];
            As[row][grp * 4 + 0] = sv.x;
            As[row][grp * 4 + 1] = sv.y;
            As[row][grp * 4 + 2] = sv.z;
            As[row][grp * 4 + 3] = sv.w;

            // neighbor mean: 25 coalesced float4 loads per slot
            const float4* p = xn + (m0 + row) * (NNEI * FDIM / 4) + grp;
            float ax = 0.f, ay = 0.f, az = 0.f, aw = 0.f;
            #pragma unroll
            for (int n = 0; n < NNEI; ++n) {
                float4 t = p[(size_t)n * (FDIM / 4)];
                ax += t.x; ay += t.y; az += t.z; aw += t.w;
            }
            An[row][grp * 4 + 0] = ax * inv_n;
            An[row][grp * 4 + 1] = ay * inv_n;
            An[row][grp * 4 + 2] = az * inv_n;
            An[row][grp * 4 + 3] = aw * inv_n;
        }
    }
    __syncthreads();

    // ---------------- Phase 2: WMMA GEMMs (f32, 16x16x4) ----------------
    const int wid     = tid >> 5;        // 8 waves: column tile [16*wid, 16*wid+16)
    const int lane    = tid & 31;
    const int lane_lo = lane & 15;       // N index (B/C/D) or M index (A)
    const int lane_hi = lane >> 4;       // K-half selector
    const int n0      = wid * 16;

    #pragma unroll
    for (int half = 0; half < 2; ++half) {
        const float (*A)[LDA] = half ? An : As;
        const float* __restrict__ W = half ? w_neigh : w_self;
        const int colbase = half * DHALF;

        v8f c = {};
        #pragma unroll 8
        for (int kk = 0; kk < FDIM / 4; ++kk) {
            const int k = kk * 4 + 2 * lane_hi;   // K for VGPR0; VGPR1 = k+1
            v2f a, b;
            a.x = A[lane_lo][k];
            a.y = A[lane_lo][k + 1];
            b.x = W[(size_t)k       * DHALF + n0 + lane_lo];
            b.y = W[(size_t)(k + 1) * DHALF + n0 + lane_lo];
            c = __builtin_amdgcn_wmma_f32_16x16x4_f32(
                    /*neg_a=*/false, a, /*neg_b=*/false, b,
                    /*c_mod=*/(short)0, c, /*reuse_a=*/false, /*reuse_b=*/false);
        }

        const int   col = colbase + n0 + lane_lo;
        const float bv  = bias[col];
        #pragma unroll
        for (int v = 0; v < 8; ++v) {
            const size_t row = m0 + (size_t)(v + 8 * lane_hi);
            float val = c[v] + bv;
            out[row * DOUT + col] = val > 0.f ? val : 0.f;
        }
    }
}

extern "C" void kernel_launch(void* const* d_in, const int* in_sizes, int n_in,
                              void* d_out, int out_size, void* d_ws, size_t ws_size,
                              hipStream_t stream) {
    const float* x_self  = (const float*)d_in[0];
    const float* x_neigh = (const float*)d_in[1];
    const float* w_neigh = (const float*)d_in[2];
    const float* w_self  = (const float*)d_in[3];
    const float* bias    = (const float*)d_in[4];
    float* out           = (float*)d_out;

    const int rows = in_sizes[0] / FDIM;        // B*H = 81920
    const int grid = rows / MROWS;              // 5120 blocks
    sage_wmma_kernel<<<grid, 256, 0, stream>>>(x_self, x_neigh, w_neigh,
                                               w_self, bias, out);
    (void)d_ws; (void)ws_size; (void)n_in; (void)out_size;
}